// GCNLayer_41068477285088
// MI455X (gfx1250) — compile-verified
//
#include <hip/hip_runtime.h>

// ---------------------------------------------------------------------------
// Batched COO SpMM for gfx1250 (MI455X).
//   out[k][dst[e]] += w[e] * embeds[k][src[e]]   (K=4, D=64, N=100k, E=1.6M)
//
// Pipeline: histogram dst -> shuffle-scan to CSR offsets -> scatter (src,w)
// into per-dst bins -> per-node register accumulation + write-once stores.
// ~3.2M u32 atomics total instead of 410M f32 atomics.
//
// CDNA5 features: async global->LDS edge staging (global_load_async_to_lds_b128
// + s_wait_asynccnt), wave32 shuffle scan, non-temporal loads of the one-shot
// bin stream and non-temporal output stores (protects L2-resident embeds).
// ---------------------------------------------------------------------------

#define TILE_E 256

typedef float v2f __attribute__((ext_vector_type(2)));
typedef float v4f __attribute__((ext_vector_type(4)));

// CDNA5 async global->LDS copy: each lane copies 16 bytes.
__device__ __forceinline__ void async_copy_b128(unsigned lds_byte_addr,
                                                const void* gaddr) {
  asm volatile("global_load_async_to_lds_b128 %0, %1, off"
               :: "v"(lds_byte_addr), "v"(gaddr)
               : "memory");
}

__device__ __forceinline__ void async_wait0() {
  asm volatile("s_wait_asynccnt 0" ::: "memory");
}

__device__ __forceinline__ unsigned lds_off(const void* p) {
  // Generic pointers to LDS carry the LDS byte offset in the low 32 bits.
  return (unsigned)(size_t)p;
}

// ---------------------------------------------------------------------------
// Utility: zero int array.
// ---------------------------------------------------------------------------
__global__ void zero_i32(int* __restrict__ p, int n) {
  int i = blockIdx.x * blockDim.x + threadIdx.x;
  if (i < n) p[i] = 0;
}

// ---------------------------------------------------------------------------
// Pass 1: histogram of destination nodes. dst chunk async-staged to LDS.
// ---------------------------------------------------------------------------
__global__ __launch_bounds__(256) void hist_kernel(const int* __restrict__ eidx,
                                                   int E,
                                                   int* __restrict__ counts) {
  __shared__ __align__(16) int sdst[TILE_E];
  const int tid = threadIdx.x;
  const int e0  = blockIdx.x * TILE_E;
  const int cnt = min(TILE_E, E - e0);

  if (cnt == TILE_E) {
    const int t16 = tid * 16;
    if (t16 < TILE_E * 4)
      async_copy_b128(lds_off(sdst) + t16, (const char*)(eidx + E + e0) + t16);
    async_wait0();
  } else {
    for (int i = tid; i < cnt; i += 256) sdst[i] = eidx[E + e0 + i];
  }
  __syncthreads();

  if (tid < cnt) atomicAdd(&counts[sdst[tid]], 1);
}

// ---------------------------------------------------------------------------
// Pass 2: exclusive scan over N counts -> offsets[0..N], cursors[0..N-1].
// Single 1024-thread block; per chunk: wave32 shuffle scans (no barriers)
// + one 32-element cross-wave scan => 4 barriers per 1024 elements.
// counts may alias cursors (per-thread read-then-write of same index).
// ---------------------------------------------------------------------------
__global__ __launch_bounds__(1024) void scan_kernel(const int* counts,
                                                    int* offsets,
                                                    int* cursors,
                                                    int N) {
  __shared__ int wsum[32];
  __shared__ int carry;
  const int tid  = threadIdx.x;
  const int lane = tid & 31;
  const int wv   = tid >> 5;
  if (tid == 0) carry = 0;
  __syncthreads();

  for (int base = 0; base < N; base += 1024) {
    const int i = base + tid;
    const int v = (i < N) ? counts[i] : 0;

    // Intra-wave inclusive scan (wave32 shuffles, no barriers).
    int incl = v;
#pragma unroll
    for (int d = 1; d < 32; d <<= 1) {
      int t = __shfl_up(incl, d, 32);
      if (lane >= d) incl += t;
    }
    if (lane == 31) wsum[wv] = incl;
    __syncthreads();

    // Wave 0 scans the 32 wave totals (inclusive).
    if (wv == 0) {
      int s = wsum[lane];
#pragma unroll
      for (int d = 1; d < 32; d <<= 1) {
        int t = __shfl_up(s, d, 32);
        if (lane >= d) s += t;
      }
      wsum[lane] = s;
    }
    __syncthreads();

    const int waveBase = (wv == 0) ? 0 : wsum[wv - 1];
    const int excl = carry + waveBase + incl - v;   // carry read before update
    if (i < N) { offsets[i] = excl; cursors[i] = excl; }
    __syncthreads();
    if (tid == 1023) carry += wsum[31];             // chunk total
    __syncthreads();
  }
  if (tid == 0) offsets[N] = carry;                 // == E
}

// ---------------------------------------------------------------------------
// Pass 3: scatter (src, w) into per-dst bins. Edge chunk async-staged to LDS.
// ---------------------------------------------------------------------------
__global__ __launch_bounds__(256) void scatter_kernel(
    const float* __restrict__ ew, const int* __restrict__ eidx, int E,
    int* __restrict__ cursors, int* __restrict__ bin_src,
    float* __restrict__ bin_w) {
  __shared__ __align__(16) float sw[TILE_E];
  __shared__ __align__(16) int   ssrc[TILE_E];
  __shared__ __align__(16) int   sdst[TILE_E];

  const int tid = threadIdx.x;
  const int e0  = blockIdx.x * TILE_E;
  const int cnt = min(TILE_E, E - e0);

  if (cnt == TILE_E) {
    const int t16 = tid * 16;
    if (t16 < TILE_E * 4) {
      async_copy_b128(lds_off(sw)   + t16, (const char*)(ew + e0)       + t16);
      async_copy_b128(lds_off(ssrc) + t16, (const char*)(eidx + e0)     + t16);
      async_copy_b128(lds_off(sdst) + t16, (const char*)(eidx + E + e0) + t16);
    }
    async_wait0();
  } else {
    for (int i = tid; i < cnt; i += 256) {
      sw[i]   = ew[e0 + i];
      ssrc[i] = eidx[e0 + i];
      sdst[i] = eidx[E + e0 + i];
    }
  }
  __syncthreads();

  if (tid < cnt) {
    const int p = atomicAdd(&cursors[sdst[tid]], 1);
    bin_src[p] = ssrc[tid];
    bin_w[p]   = sw[tid];
  }
}

// ---------------------------------------------------------------------------
// Pass 4: per-node accumulation. One wave32 per node. Lane mapping:
//   l  = lane & 15 -> owns feature quad (16 lanes x float4 = D = 64)
//   kk = lane >> 4 -> owns batch pair {2kk, 2kk+1}
// => 2 b128 gathers per edge per lane, float4 accumulators, NT epilogue.
// ---------------------------------------------------------------------------
__global__ __launch_bounds__(256) void accum_kernel(
    const float* __restrict__ embeds, const int* __restrict__ offsets,
    const int* __restrict__ bin_src, const float* __restrict__ bin_w,
    float* __restrict__ out, int N) {
  const int wave = threadIdx.x >> 5;
  const int lane = threadIdx.x & 31;
  const int node = blockIdx.x * 8 + wave;
  if (node >= N) return;

  const int beg = offsets[node];
  const int end = offsets[node + 1];

  const int l  = lane & 15;
  const int kk = lane >> 4;
  const size_t co    = (size_t)(l * 4);
  const size_t rowk0 = (size_t)(2 * kk) * N;
  const size_t rowk1 = (size_t)(2 * kk + 1) * N;

  v4f a0 = {0.f, 0.f, 0.f, 0.f};
  v4f a1 = {0.f, 0.f, 0.f, 0.f};

  for (int j = beg; j < end; ++j) {
    // Bin stream is read exactly once -> non-temporal (don't evict embeds).
    const int   s = __builtin_nontemporal_load(&bin_src[j]);
    const float w = __builtin_nontemporal_load(&bin_w[j]);
    const v4f h0 = *(const v4f*)(embeds + (rowk0 + (size_t)s) * 64 + co);
    const v4f h1 = *(const v4f*)(embeds + (rowk1 + (size_t)s) * 64 + co);
    a0 += w * h0;
    a1 += w * h1;
  }

  // Write-once output: non-temporal so it doesn't evict embeds from L2.
  __builtin_nontemporal_store(a0, (v4f*)(out + (rowk0 + (size_t)node) * 64 + co));
  __builtin_nontemporal_store(a1, (v4f*)(out + (rowk1 + (size_t)node) * 64 + co));
}

// ---------------------------------------------------------------------------
// Fallback path (workspace too small): zero + atomic scatter.
// ---------------------------------------------------------------------------
__global__ void spmm_zero(float* __restrict__ out, int n) {
  int i = (blockIdx.x * blockDim.x + threadIdx.x) * 4;
  if (i + 3 < n) {
    *reinterpret_cast<float4*>(out + i) = make_float4(0.f, 0.f, 0.f, 0.f);
  } else {
    for (int j = i; j < n; ++j) out[j] = 0.f;
  }
}

__global__ __launch_bounds__(256) void spmm_scatter(
    const float* __restrict__ ew, const int* __restrict__ eidx,
    const float* __restrict__ embeds, float* __restrict__ out, int E, int N) {
  __shared__ __align__(16) float sw[TILE_E];
  __shared__ __align__(16) int   ssrc[TILE_E];
  __shared__ __align__(16) int   sdst[TILE_E];

  const int tid = threadIdx.x;
  const int e0  = blockIdx.x * TILE_E;
  const int cnt = min(TILE_E, E - e0);

  if (cnt == TILE_E) {
    const int t16 = tid * 16;
    if (t16 < TILE_E * 4) {
      async_copy_b128(lds_off(sw)   + t16, (const char*)(ew + e0)       + t16);
      async_copy_b128(lds_off(ssrc) + t16, (const char*)(eidx + e0)     + t16);
      async_copy_b128(lds_off(sdst) + t16, (const char*)(eidx + E + e0) + t16);
    }
    async_wait0();
  } else {
    for (int i = tid; i < cnt; i += 256) {
      sw[i]   = ew[e0 + i];
      ssrc[i] = eidx[e0 + i];
      sdst[i] = eidx[E + e0 + i];
    }
  }
  __syncthreads();

  const int wave = tid >> 5;
  const int lane = tid & 31;
  const size_t co = (size_t)(lane * 2);
  for (int i = wave; i < cnt; i += 8) {
    const int s = ssrc[i], d = sdst[i];
    const float w = sw[i];
#pragma unroll
    for (int k = 0; k < 4; ++k) {
      const v2f h = *(const v2f*)(embeds + (size_t)(k * N + s) * 64 + co);
      float* o = out + (size_t)(k * N + d) * 64 + co;
      atomicAdd(o,     w * h.x);
      atomicAdd(o + 1, w * h.y);
    }
  }
}

// ---------------------------------------------------------------------------
// Launcher. Inputs (setup_inputs order):
//   d_in[0] edge_weight f32[E], d_in[1] embeds f32[K*N*D],
//   d_in[2] edge_index i32[2*E], d_in[3] num_nodes (unused).
// Workspace (binned path): offsets[N+1] | cursors[N] | bin_src[E] | bin_w[E]
// ---------------------------------------------------------------------------
extern "C" void kernel_launch(void* const* d_in, const int* in_sizes, int n_in,
                              void* d_out, int out_size, void* d_ws, size_t ws_size,
                              hipStream_t stream) {
  (void)n_in;

  const float* ew     = (const float*)d_in[0];
  const float* embeds = (const float*)d_in[1];
  const int*   eidx   = (const int*)d_in[2];
  float*       out    = (float*)d_out;

  const int E = in_sizes[0];
  const int N = in_sizes[1] / (4 * 64);   // K=4, D=64
  const int nchunks = (E + TILE_E - 1) / TILE_E;

  const size_t need = ((size_t)2 * N + 1 + 2 * (size_t)E) * sizeof(int);

  if (d_ws != nullptr && ws_size >= need) {
    int*   offsets = (int*)d_ws;
    int*   cursors = offsets + (N + 1);
    int*   bin_src = cursors + N;
    float* bin_w   = (float*)(bin_src + E);

    zero_i32<<<(N + 255) / 256, 256, 0, stream>>>(cursors, N);
    hist_kernel<<<nchunks, 256, 0, stream>>>(eidx, E, cursors);
    scan_kernel<<<1, 1024, 0, stream>>>(cursors, offsets, cursors, N);
    scatter_kernel<<<nchunks, 256, 0, stream>>>(ew, eidx, E, cursors, bin_src, bin_w);
    accum_kernel<<<(N + 7) / 8, 256, 0, stream>>>(embeds, offsets, bin_src, bin_w, out, N);
  } else {
    spmm_zero<<<(out_size / 4 + 255) / 256, 256, 0, stream>>>(out, out_size);
    spmm_scatter<<<nchunks, 256, 0, stream>>>(ew, eidx, embeds, out, E, N);
  }
}